// RiemannianManifold_76965813944498
// MI455X (gfx1250) — compile-verified
//
#include <hip/hip_runtime.h>
#include <math.h>

// CDNA5 / gfx1250, wave32. Uses V_WMMA_F32_16X16X4_F32 (fp32 matrix core).
typedef __attribute__((ext_vector_type(2))) float v2f;
typedef __attribute__((ext_vector_type(8))) float v8f;

#define WAVES_PER_WG 4
// padded LDS strides (floats) to spread across the 64 banks; kept even so
// float2 (ds_load_b64) accesses stay 8-byte aligned.
#define MIDS 18
#define DXS  18
#define HS   34
#define W1S  18
#define W2S  34

#define DIAG_MIN 0.316f
#define DIAG_MAX 3.16f

__device__ __forceinline__ v8f wmma_f32_16x16x4(v2f a, v2f b, v8f c) {
  // 8 args: (neg_a, A, neg_b, B, c_mod, C, reuse_a, reuse_b)
  return __builtin_amdgcn_wmma_f32_16x16x4_f32(false, a, false, b, (short)0, c,
                                               false, false);
}

// silu via hardware v_rcp_f32 (avoids the IEEE div_scale/fma chain)
__device__ __forceinline__ float silu_fast(float x) {
  return x * __builtin_amdgcn_rcpf(1.0f + __expf(-x));
}

// softplus via native v_exp_f32 / v_log_f32; safe under the [0.316,3.16]
// clamp: overflow -> inf -> clamped hi; underflow -> log(1)=0 -> clamped lo.
__device__ __forceinline__ float softplus_fast(float x) {
  return __logf(1.0f + __expf(x));
}

__global__ __launch_bounds__(128) void RiemannianManifold_kernel(
    const float* __restrict__ x1, const float* __restrict__ x2,
    const float* __restrict__ W1, const float* __restrict__ b1,
    const float* __restrict__ W2, const float* __restrict__ b2,
    float* __restrict__ out) {
  __shared__ float sW1[32 * W1S];
  __shared__ float sW2[256 * W2S];
  __shared__ float sb1[32];
  __shared__ float sb2[256];
  __shared__ float smid[WAVES_PER_WG][16 * MIDS];
  __shared__ float sdx[WAVES_PER_WG][16 * DXS];
  __shared__ float sh[WAVES_PER_WG][16 * HS];

  const int tid = threadIdx.x;
  const int lane = tid & 31;
  const int wv = tid >> 5;
  const int lo = lane & 15;  // A-frag row / B-frag col / C-layout N index
  const int hi = lane >> 4;  // half-wave select
  const int koff = 2 * hi;   // K offset within a K=4 step (ISA f32 A/B layout)

  // ---------------- stage weights/biases into LDS (padded) ----------------
  for (int idx = tid; idx < 256 * 32; idx += 128)
    sW2[(idx >> 5) * W2S + (idx & 31)] = W2[idx];
  for (int idx = tid; idx < 32 * 16; idx += 128)
    sW1[(idx >> 4) * W1S + (idx & 15)] = W1[idx];
  for (int idx = tid; idx < 256; idx += 128) sb2[idx] = b2[idx];
  if (tid < 32) sb1[tid] = b1[tid];

  // ---------------- wave-tile decode: (b, n, 16 consecutive m) ------------
  const int gw = blockIdx.x * WAVES_PER_WG + wv;  // 0..8191
  const int mblk = gw & 15;
  const int n = (gw >> 4) & 255;
  const int b = gw >> 12;

  // ---------------- per-pair mid / dx into LDS ----------------------------
  {
    const int r = lo;  // pair row 0..15
    const int m = mblk * 16 + r;
    const float* p1 = x1 + ((b * 256 + n) * 16 + 8 * hi);
    const float* p2 = x2 + ((b * 256 + m) * 16 + 8 * hi);
    float* dm = &smid[wv][r * MIDS + 8 * hi];
    float* dd = &sdx[wv][r * DXS + 8 * hi];
#pragma unroll
    for (int k = 0; k < 8; ++k) {
      const float a1 = p1[k], a2 = p2[k];
      dm[k] = 0.5f * (a1 + a2);
      dd[k] = a2 - a1;
    }
  }
  __syncthreads();

  // ---------------- layer 1: h = silu(mid @ W1^T + b1) --------------------
  // A fragments: lane lo holds mid[lo][4s + koff + {0,1}]
  v2f am[4];
#pragma unroll
  for (int s = 0; s < 4; ++s)
    am[s] = *(const v2f*)&smid[wv][lo * MIDS + 4 * s + koff];

#pragma unroll
  for (int t = 0; t < 2; ++t) {  // hid N-tiles j = 16t .. 16t+15
    v8f c = {};
#pragma unroll
    for (int s = 0; s < 4; ++s) {
      // B[k][j] = W1[j][k]; lane lo supplies column j = 16t+lo
      v2f bf = *(const v2f*)&sW1[(t * 16 + lo) * W1S + 4 * s + koff];
      c = wmma_f32_16x16x4(am[s], bf, c);
    }
    const int j = t * 16 + lo;
    const float bias = sb1[j];
#pragma unroll
    for (int v = 0; v < 8; ++v) {  // C/D layout: pair r = v + 8*hi, col = j
      const int r = v + 8 * hi;
      sh[wv][r * HS + j] = silu_fast(c[v] + bias);
    }
  }
  __syncthreads();

  // ---------------- layer 2 + L construction + v = L^T dx -----------------
  // A = h (16 pairs x 32 hid): lane lo holds h[lo][4s + koff + {0,1}]
  v2f ah[8];
#pragma unroll
  for (int s = 0; s < 8; ++s)
    ah[s] = *(const v2f*)&sh[wv][lo * HS + 4 * s + koff];

  v8f vacc = {};  // vacc[v] = v_j for pair (v + 8*hi), j = lo

  for (int t = 0; t < 16; ++t) {  // N-tile t == row t of L, all 16 pairs
    v8f c = {};
#pragma unroll
    for (int s = 0; s < 8; ++s) {
      // B[k][j] = W2[j][k]; lane lo supplies column j = 16t+lo
      v2f bf = *(const v2f*)&sW2[(t * 16 + lo) * W2S + 4 * s + koff];
      c = wmma_f32_16x16x4(ah[s], bf, c);
    }
    const float bias = sb2[t * 16 + lo];
#pragma unroll
    for (int v = 0; v < 8; ++v) {
      const int r = v + 8 * hi;  // pair index
      float lv = c[v] + bias;    // raw[r][t*16 + lo]
      if (lo == t) {             // diagonal: +I, softplus, clamp
        lv = fminf(fmaxf(softplus_fast(lv + 1.0f), DIAG_MIN), DIAG_MAX);
      } else if (lo > t) {       // strict upper: tril mask
        lv = 0.0f;
      }
      // v[j] += L[t][j] * dx[t]   (j = lo)
      vacc[v] += lv * sdx[wv][r * DXS + t];
    }
  }

  // ---------------- dist = sqrt(max(sum_j v_j^2, 1e-6)) -------------------
  const int outbase = (b * 256 + n) * 256 + mblk * 16;
#pragma unroll
  for (int v = 0; v < 8; ++v) {
    float sq = vacc[v] * vacc[v];
    sq += __shfl_xor(sq, 1, 32);  // reduce across the 16-lane half
    sq += __shfl_xor(sq, 2, 32);
    sq += __shfl_xor(sq, 4, 32);
    sq += __shfl_xor(sq, 8, 32);
    if (lo == v) out[outbase + v + 8 * hi] = sqrtf(fmaxf(sq, 1e-6f));
  }
}

extern "C" void kernel_launch(void* const* d_in, const int* in_sizes, int n_in,
                              void* d_out, int out_size, void* d_ws,
                              size_t ws_size, hipStream_t stream) {
  (void)in_sizes; (void)n_in; (void)out_size; (void)d_ws; (void)ws_size;
  const float* x1 = (const float*)d_in[0];
  const float* x2 = (const float*)d_in[1];
  const float* W1 = (const float*)d_in[2];
  const float* b1 = (const float*)d_in[3];
  const float* W2 = (const float*)d_in[4];
  const float* b2 = (const float*)d_in[5];
  float* out = (float*)d_out;

  // 8192 wave-tiles (B=2 * N=256 * M/16=16), 4 waves per 128-thread block
  const int blocks = 8192 / WAVES_PER_WG;
  RiemannianManifold_kernel<<<blocks, 128, 0, stream>>>(x1, x2, W1, b1, W2, b2,
                                                        out);
}